// SphericalBesselBasis_27994596835747
// MI455X (gfx1250) — compile-verified
//
#include <hip/hip_runtime.h>
#include <hip/hip_bf16.h>

// ---------------------------------------------------------------------------
// DimeNet spherical Bessel basis for MI455X (gfx1250).
//   Pass 1: per-edge radial table [E,42] = env(d) * N_ln * j_l(z_ln * d)
//           (84 MB -> resident in the 192 MB L2; cuts transcendentals 4x)
//   Pass 2: 16-angle tiles; cbf via two chained V_WMMA_F32_16X16X4_F32
//           (Legendre as [16x8]x[8x16] monomial matmul), B64 gather-mul-store.
// ---------------------------------------------------------------------------

#define L_SPH 7
#define N_RAD 6
#define LN 42
#define LN2 21  // float2 columns per row
#define CUTOFF_F 5.0f
// envelope p = ENV_EXP + 1 = 6
#define ENV_A (-28.0f)
#define ENV_B (48.0f)
#define ENV_C (-21.0f)

typedef __attribute__((ext_vector_type(2))) float v2f;
typedef __attribute__((ext_vector_type(4))) float v4f;
typedef __attribute__((ext_vector_type(8))) float v8f;

// Y_l0 normalizers sqrt((2l+1)/4pi)
#define S0 0.28209479177387814f
#define S1 0.48860251190291992f
#define S2 0.63078313050504009f
#define S3 0.74635266518023078f
#define S4 0.84628437532163443f
#define S5 0.93560257962738882f
#define S6 1.01710723628205458f

// B-operand of V_WMMA_F32_16X16X4_F32, pre-swizzled per lane.
// Assumed layout mirrors the ISA A-table: lane = N column (mod 16),
// v0 = K{0|2}, v1 = K{1|3} per lane half.  BP0 = K0..K3, BP1 = K4..K7.
// BP*[lane] = (B[Klo][lane&15], B[Khi][lane&15]) with Klo/Khi per lane half.
__constant__ __align__(16) float BP0[64] = {
  // lanes 0..15: (K0[m], K1[m])
  S0, 0.f,            0.f, S1,            -0.5f*S2, 0.f,    0.f, -1.5f*S3,
  0.375f*S4, 0.f,     0.f, 1.875f*S5,     -0.3125f*S6, 0.f, 0.f, 0.f,
  0,0, 0,0, 0,0, 0,0, 0,0, 0,0, 0,0, 0,0,
  // lanes 16..31: (K2[m], K3[m])
  0.f, 0.f,           0.f, 0.f,           1.5f*S2, 0.f,     0.f, 2.5f*S3,
  -3.75f*S4, 0.f,     0.f, -8.75f*S5,     6.5625f*S6, 0.f,  0.f, 0.f,
  0,0, 0,0, 0,0, 0,0, 0,0, 0,0, 0,0, 0,0
};
__constant__ __align__(16) float BP1[64] = {
  // lanes 0..15: (K4[m], K5[m])
  0.f, 0.f,           0.f, 0.f,           0.f, 0.f,         0.f, 0.f,
  4.375f*S4, 0.f,     0.f, 7.875f*S5,     -19.6875f*S6, 0.f, 0.f, 0.f,
  0,0, 0,0, 0,0, 0,0, 0,0, 0,0, 0,0, 0,0,
  // lanes 16..31: (K6[m], K7[m])  (row K7 is zero)
  0.f, 0.f,           0.f, 0.f,           0.f, 0.f,         0.f, 0.f,
  0.f, 0.f,           0.f, 0.f,           14.4375f*S6, 0.f, 0.f, 0.f,
  0,0, 0,0, 0,0, 0,0, 0,0, 0,0, 0,0, 0,0
};

__constant__ float SLCONST[7] = { S0, S1, S2, S3, S4, S5, S6 };

__device__ __forceinline__ float envelope(float d) {
  float invd = 1.0f / d;
  float d2 = d * d;
  float d5 = d2 * d2 * d;
  float env = invd + d5 * (ENV_A + d * (ENV_B + d * ENV_C));
  return (d < 1.0f) ? env : 0.0f;
}

__device__ __forceinline__ void radial42(float d, const float* __restrict__ zeros,
                                         const float* __restrict__ norms, float env,
                                         float* __restrict__ out42) {
#pragma unroll
  for (int l = 0; l < L_SPH; ++l) {
#pragma unroll
    for (int n = 0; n < N_RAD; ++n) {
      float x = zeros[l * N_RAD + n] * d;
      float s, c;
      __sincosf(x, &s, &c);
      float invx = 1.0f / x;
      float j0 = s * invx;
      float jl;
      if (l == 0) {
        jl = j0;
      } else {
        float j1 = (s * invx - c) * invx;
        float jm2 = j0, jm1 = j1;
#pragma unroll
        for (int k = 2; k <= l; ++k) {
          float t = (2.0f * (float)k - 1.0f) * invx * jm1 - jm2;
          jm2 = jm1;
          jm1 = t;
        }
        jl = jm1;
      }
      out42[l * N_RAD + n] = env * norms[l * N_RAD + n] * jl;
    }
  }
}

// ---------------- Pass 1: per-edge radial table, LDS-staged B64 stores -------
__global__ void __launch_bounds__(256)
edge_radial_kernel(const float* __restrict__ distance,
                   const float* __restrict__ zeros,
                   const float* __restrict__ norms,
                   float* __restrict__ table, int E) {
  __shared__ __align__(16) float st[256 * LN];  // 43 KB of 320 KB WGP LDS
  int tid = threadIdx.x;
  int e = blockIdx.x * 256 + tid;

  float d = 2.0f;  // OOB lanes: env==0, radial finite
  if (e < E) d = distance[e] * (1.0f / CUTOFF_F);
  float env = envelope(d);

  float vals[LN];
  radial42(d, zeros, norms, env, vals);

  v2f* st2 = (v2f*)st;
#pragma unroll
  for (int j = 0; j < LN2; ++j)
    st2[tid * LN2 + j] = (v2f){vals[2 * j], vals[2 * j + 1]};
  __syncthreads();

  v2f* t2 = (v2f*)table;
  size_t base2 = (size_t)blockIdx.x * 256 * LN2;
  size_t limit2 = (size_t)E * LN2;
#pragma unroll 7
  for (int i = tid; i < 256 * LN2; i += 256) {  // 21 iterations
    size_t g = base2 + (size_t)i;
    if (g < limit2) t2[g] = st2[i];
  }
}

// ---------------- Pass 2: 16-angle tiles, cbf via WMMA, B64 gather-mul-store -
// Block = 256 threads = 8 waves; each wave owns a 16-angle tile.
__global__ void __launch_bounds__(256)
angle_outer_kernel(const float* __restrict__ angle,
                   const int* __restrict__ kj,
                   const float* __restrict__ table,
                   float* __restrict__ out, int A) {
  // scbf[wave][N=col(l)][M=row(angle)] so each lane spills its D column
  // (8 contiguous rows) with two ds_store_b128.
  __shared__ __align__(16) float scbf[8][16][16];  // 8 KB
  __shared__ int skj2[8][16];                      // row offsets in float2 units

  const int lane = threadIdx.x & 31;
  const int wave = threadIdx.x >> 5;
  const int m = lane & 15;  // angle row for A, column N for B/D
  const bool hi = lane >= 16;
  const int aBase = (blockIdx.x * 8 + wave) * 16;

  const int am = aBase + m;
  const bool valid = am < A;
  float th = valid ? angle[am] : 0.0f;
  int kjv = valid ? kj[am] : 0;
  if (lane < 16) {
    skj2[wave][m] = kjv * LN2;
    if (valid) __builtin_prefetch(table + (size_t)kjv * LN, 0, 3);  // global_prefetch_b8
  }

  // A operand: monomial powers of cos(theta). Lane = M (angle row);
  // v0 holds K{0|2}, v1 holds K{1|3} per lane half (ISA 16x4 f32 A layout).
  float c1 = __cosf(th);
  float c2 = c1 * c1, c3 = c2 * c1, c4 = c2 * c2, c5 = c4 * c1, c6 = c4 * c2;
  v2f av0 = hi ? (v2f){c2, c3} : (v2f){1.0f, c1};   // K=2,3 | K=0,1
  v2f av1 = hi ? (v2f){c6, 0.0f} : (v2f){c4, c5};   // K=6,7 | K=4,5

  // B operand: lane-indexed pre-swizzled coefficient pairs (two B64 loads).
  v2f bv0 = ((const v2f*)BP0)[lane];
  v2f bv1 = ((const v2f*)BP1)[lane];

  // cbf[16x16] = PW[16x8] x BMAT[8x16] as two chained K=4 f32 WMMAs.
  v8f acc = {};
  acc = __builtin_amdgcn_wmma_f32_16x16x4_f32(false, av0, false, bv0,
                                              (short)0, acc, false, false);
  acc = __builtin_amdgcn_wmma_f32_16x16x4_f32(false, av1, false, bv1,
                                              (short)0, acc, false, false);

  // D layout: VGPR r, lanes<16 -> M=r, lanes>=16 -> M=r+8; N = lane&15.
  // Lane owns column N=m, rows 0..7 (lo half) or 8..15 (hi half): 2x B128.
  float* p = &scbf[wave][m][hi ? 8 : 0];
  *(v4f*)(p + 0) = (v4f){acc[0], acc[1], acc[2], acc[3]};
  *(v4f*)(p + 4) = (v4f){acc[4], acc[5], acc[6], acc[7]};
  __builtin_amdgcn_wave_barrier();  // exchange is wave-local; DS is in-order

  // 16 angles x 21 float2 = 336 pairs; 10 uniform iterations + 16-lane tail.
  // Pair (2j, 2j+1) never straddles an l-boundary -> single cb per pair.
  const v2f* t2 = (const v2f*)table;
  v2f* o2 = (v2f*)out;
#pragma unroll
  for (int k = 0; k < 10; ++k) {
    int i = lane + 32 * k;
    int arow = i / LN2;
    int c2i = i - arow * LN2;
    int a = aBase + arow;
    if (a < A) {
      float cb = scbf[wave][c2i / 3][arow];
      v2f rv = t2[(size_t)skj2[wave][arow] + (size_t)c2i];
      o2[(size_t)a * LN2 + (size_t)c2i] = (v2f){rv.x * cb, rv.y * cb};
    }
  }
  if (lane < 16) {  // tail: i = 320..335
    int i = lane + 320;
    int arow = i / LN2;
    int c2i = i - arow * LN2;
    int a = aBase + arow;
    if (a < A) {
      float cb = scbf[wave][c2i / 3][arow];
      v2f rv = t2[(size_t)skj2[wave][arow] + (size_t)c2i];
      o2[(size_t)a * LN2 + (size_t)c2i] = (v2f){rv.x * cb, rv.y * cb};
    }
  }
}

// ---------------- Fallback: fused one-pass (only if d_ws is too small) -------
__global__ void __launch_bounds__(256)
fused_kernel(const float* __restrict__ distance,
             const float* __restrict__ angle,
             const int* __restrict__ kj,
             const float* __restrict__ zeros,
             const float* __restrict__ norms,
             float* __restrict__ out, int A) {
  int a = blockIdx.x * 256 + threadIdx.x;
  if (a >= A) return;
  float d = distance[kj[a]] * (1.0f / CUTOFF_F);
  float env = envelope(d);
  float vals[LN];
  radial42(d, zeros, norms, env, vals);

  float c = __cosf(angle[a]);
  float P[L_SPH];
  P[0] = 1.0f;
  P[1] = c;
#pragma unroll
  for (int l = 2; l < L_SPH; ++l)
    P[l] = ((2.0f * l - 1.0f) * c * P[l - 1] - (l - 1.0f) * P[l - 2]) / (float)l;

  v2f* o2 = (v2f*)(out + (size_t)a * LN);
#pragma unroll
  for (int j = 0; j < LN2; ++j) {
    float cb = SLCONST[j / 3] * P[j / 3];
    o2[j] = (v2f){vals[2 * j] * cb, vals[2 * j + 1] * cb};
  }
}

extern "C" void kernel_launch(void* const* d_in, const int* in_sizes, int n_in,
                              void* d_out, int out_size, void* d_ws, size_t ws_size,
                              hipStream_t stream) {
  const float* distance = (const float*)d_in[0];
  const float* angle    = (const float*)d_in[1];
  const int*   kj       = (const int*)d_in[2];  // angle_index[0] = first A entries
  const float* zeros    = (const float*)d_in[3];
  const float* norms    = (const float*)d_in[4];
  float* out = (float*)d_out;

  const int E = in_sizes[0];
  const int A = in_sizes[1];
  const size_t table_bytes = (size_t)E * LN * sizeof(float);

  if (ws_size >= table_bytes) {
    float* table = (float*)d_ws;
    int g1 = (E + 255) / 256;
    edge_radial_kernel<<<g1, 256, 0, stream>>>(distance, zeros, norms, table, E);
    int g2 = (A + 127) / 128;  // 8 waves * 16 angles per block
    angle_outer_kernel<<<g2, 256, 0, stream>>>(angle, kj, table, out, A);
  } else {
    int g = (A + 255) / 256;
    fused_kernel<<<g, 256, 0, stream>>>(distance, angle, kj, zeros, norms, out, A);
  }
}